// QuantumRNNCell_43800076484690
// MI455X (gfx1250) — compile-verified
//
#include <hip/hip_runtime.h>

#define NQ  6
#define DIM 64

typedef __attribute__((ext_vector_type(2))) float v2f;
typedef __attribute__((ext_vector_type(8))) float v8f;

struct c2 { float r, i; };
__device__ __forceinline__ c2 cmul(c2 a, c2 b) { return c2{ a.r*b.r - a.i*b.i, a.r*b.i + a.i*b.r }; }
__device__ __forceinline__ c2 cadd(c2 a, c2 b) { return c2{ a.r + b.r, a.i + b.i }; }

// ---------------------------------------------------------------------------
// Kernel 1: 6-qubit statevector simulation, one thread per batch item.
// State (64 complex) lives entirely in VGPRs (all indices constant after
// unrolling). Writes PauliZ expvals padded to stride 8 (k=6,7 zeroed so the
// GEMM kernel's zero-padded K-fragment loads them directly).
// ---------------------------------------------------------------------------
__global__ __launch_bounds__(256) void qsim_kernel(const float* __restrict__ x,
                                                   const float* __restrict__ qwts,
                                                   float* __restrict__ qout, int B)
{
  int b = blockIdx.x * blockDim.x + threadIdx.x;
  if (b >= B) return;

  c2 s[DIM];
  #pragma unroll
  for (int i = 0; i < DIM; ++i) s[i] = c2{0.f, 0.f};
  s[0].r = 1.f;

  float xv[6];
  #pragma unroll
  for (int j = 0; j < 6; ++j) xv[j] = x[(size_t)b * 6 + j];

  // Input encoding: per wire w apply RX(x[w]) then RY(x[w+1]) then RZ(x[w+2]);
  // compose M = Rz * Ry * Rx once and apply a single 2x2 per wire.
  #pragma unroll
  for (int w = 0; w < NQ; ++w) {
    float tx = xv[w % 6] * 0.5f;
    float ty = xv[(w + 1) % 6] * 0.5f;
    float tz = xv[(w + 2) % 6] * 0.5f;
    float cx = __cosf(tx), sx = __sinf(tx);
    float cy = __cosf(ty), sy = __sinf(ty);
    float cz = __cosf(tz), sz = __sinf(tz);
    // T = Ry * Rx
    c2 T00{  cy * cx,  sy * sx };
    c2 T01{ -sy * cx, -cy * sx };
    c2 T10{  sy * cx, -cy * sx };
    c2 T11{  cy * cx, -sy * sx };
    c2 e0{ cz, -sz }, e1{ cz, sz };   // exp(-i tz/2), exp(+i tz/2)
    c2 M00 = cmul(e0, T00), M01 = cmul(e0, T01);
    c2 M10 = cmul(e1, T10), M11 = cmul(e1, T11);
    const int stride = 1 << (5 - w);  // wire 0 = MSB
    #pragma unroll
    for (int p = 0; p < DIM / 2; ++p) {
      int a  = ((p & ~(stride - 1)) << 1) | (p & (stride - 1));
      int bb = a | stride;
      c2 sa = s[a], sb = s[bb];
      s[a]  = cadd(cmul(M00, sa), cmul(M01, sb));
      s[bb] = cadd(cmul(M10, sa), cmul(M11, sb));
    }
  }

  // Variational layers: shared RY rotations + CNOT ring (register renames).
  #pragma unroll
  for (int l = 0; l < 3; ++l) {
    #pragma unroll
    for (int q = 0; q < NQ; ++q) {
      float th = qwts[l * NQ + q] * 0.5f;     // uniform -> scalar loads
      float c = __cosf(th), sn = __sinf(th);
      const int stride = 1 << (5 - q);
      #pragma unroll
      for (int p = 0; p < DIM / 2; ++p) {
        int a  = ((p & ~(stride - 1)) << 1) | (p & (stride - 1));
        int bb = a | stride;
        c2 sa = s[a], sb = s[bb];
        s[a]  = c2{ c * sa.r - sn * sb.r, c * sa.i - sn * sb.i };
        s[bb] = c2{ sn * sa.r + c * sb.r, sn * sa.i + c * sb.i };
      }
    }
    #pragma unroll
    for (int q = 0; q < NQ; ++q) {
      int t = (q + 1) % NQ;
      const int cm = 1 << (5 - q), tm = 1 << (5 - t);
      #pragma unroll
      for (int i = 0; i < DIM; ++i) {
        if ((i & cm) && !(i & tm)) {        // constant-folds per unrolled i
          c2 tmp = s[i]; s[i] = s[i | tm]; s[i | tm] = tmp;
        }
      }
    }
  }

  float pr[DIM];
  #pragma unroll
  for (int i = 0; i < DIM; ++i) pr[i] = s[i].r * s[i].r + s[i].i * s[i].i;
  #pragma unroll
  for (int k = 0; k < NQ; ++k) {
    float acc = 0.f;
    #pragma unroll
    for (int i = 0; i < DIM; ++i) acc += ((i >> (5 - k)) & 1) ? -pr[i] : pr[i];
    qout[(size_t)b * 8 + k] = acc;
  }
  qout[(size_t)b * 8 + 6] = 0.f;
  qout[(size_t)b * 8 + 7] = 0.f;
}

// ---------------------------------------------------------------------------
// Kernel 1b: zero-pad fc_w (H,6) -> (H,8) so all WMMA B fragments are
// unconditional aligned b64 loads (no per-lane masking / EXEC branches).
// ---------------------------------------------------------------------------
__global__ __launch_bounds__(256) void padw_kernel(const float* __restrict__ fcw,
                                                   float* __restrict__ fcw8, int H)
{
  int h = blockIdx.x * blockDim.x + threadIdx.x;
  if (h >= H) return;
  #pragma unroll
  for (int k = 0; k < 6; ++k) fcw8[(size_t)h * 8 + k] = fcw[(size_t)h * 6 + k];
  fcw8[(size_t)h * 8 + 6] = 0.f;
  fcw8[(size_t)h * 8 + 7] = 0.f;
}

// ---------------------------------------------------------------------------
// Kernel 2: out = hx + q @ fc_w^T + fc_b, streamed at the HBM roofline
// (268 MB total traffic -> ~11.5 us at 23.3 TB/s). One wave per 16-batch-row
// tile looping over H/16 column tiles. Two chained V_WMMA_F32_16X16X4_F32
// cover K=6 (k=6,7 zero-padded on both A and B via the padded tables).
//
// A layout (16x4 f32): lanes 0-15 hold K0/K1 of row M=lane in VGPR0/1,
// lanes 16-31 hold K2/K3 of row M=lane-16 -> each fragment is one b64 load
// from the stride-8 padded tables. hx/out use non-temporal hints (read-once /
// write-once streams); fc_w8 / fc_b / q stay regular-temporal (L2-resident).
// ---------------------------------------------------------------------------
__global__ __launch_bounds__(256) void resgemm_kernel(const float* __restrict__ qws,
                                                      const float* __restrict__ fcw8,
                                                      const float* __restrict__ hx,
                                                      const float* __restrict__ fcb,
                                                      float* __restrict__ out,
                                                      int B, int H)
{
  const int lane = threadIdx.x & 31;
  const int wid  = blockIdx.x * (blockDim.x >> 5) + (threadIdx.x >> 5);
  const int btiles = B >> 4;
  if (wid >= btiles) return;                 // wave-uniform: EXEC stays full
  const int brow0 = wid << 4;
  const bool hi = lane >= 16;
  const int  n  = lane & 15;
  const int  ksel = hi ? 2 : 0;              // low half: k0/k1, high: k2/k3

  // A fragments: one aligned b64 each from the stride-8 padded q table.
  const float* __restrict__ qp = qws + ((size_t)(brow0 + n) << 3) + ksel;
  const v2f A1 = *(const v2f*)(qp);          // k = ksel, ksel+1
  const v2f A2 = *(const v2f*)(qp + 4);      // k = ksel+4, ksel+5 (6,7 = 0)

  const int mbase = brow0 + (hi ? 8 : 0);

  // Per-lane streaming base pointers (advance 16 floats / tile).
  const float* __restrict__ hxp  = hx  + (size_t)mbase * H + n;
  float*       __restrict__ outp = out + (size_t)mbase * H + n;
  const float* __restrict__ wp   = fcw8 + ((size_t)n << 3) + ksel;
  const float* __restrict__ fbp  = fcb + n;

  const int HT = H >> 4;
  for (int ht = 0; ht < HT; ++ht) {
    // B fragments: B[k][col] = fc_w8[col][k]; unconditional b64 loads.
    const v2f B1 = *(const v2f*)(wp);        // k = ksel, ksel+1
    const v2f B2 = *(const v2f*)(wp + 4);    // k = ksel+4, ksel+5 (6,7 = 0)
    const float bias = fbp[0];

    v8f C;
    #pragma unroll
    for (int r = 0; r < 8; ++r)
      C[r] = __builtin_nontemporal_load(&hxp[(size_t)r * H]) + bias;

    v8f D = __builtin_amdgcn_wmma_f32_16x16x4_f32(false, A1, false, B1,
                                                  (short)0, C, false, false);
    D = __builtin_amdgcn_wmma_f32_16x16x4_f32(false, A2, false, B2,
                                              (short)0, D, false, false);

    #pragma unroll
    for (int r = 0; r < 8; ++r)
      __builtin_nontemporal_store(D[r], &outp[(size_t)r * H]);

    __builtin_prefetch(hxp + 16, 0, 3);       // gfx1250 global_prefetch_b8

    hxp  += 16;
    outp += 16;
    wp   += 128;                              // 16 columns * 8 floats
    fbp  += 16;
  }
}

extern "C" void kernel_launch(void* const* d_in, const int* in_sizes, int n_in,
                              void* d_out, int out_size, void* d_ws, size_t ws_size,
                              hipStream_t stream) {
  const float* x   = (const float*)d_in[0];   // (B, 6)
  const float* hx  = (const float*)d_in[1];   // (B, H)
  const float* qw  = (const float*)d_in[2];   // (3, 6)
  const float* fcw = (const float*)d_in[3];   // (H, 6)
  const float* fcb = (const float*)d_in[4];   // (H,)
  float* out = (float*)d_out;                 // (B, H)

  const int B = in_sizes[0] / 6;              // 32768
  const int H = in_sizes[4];                  // 1024

  float* qws  = (float*)d_ws;                 // [0, B*8) floats: padded q
  float* fcw8 = qws + (size_t)B * 8;          // [B*8, B*8+H*8): padded fc_w

  qsim_kernel<<<(B + 255) / 256, 256, 0, stream>>>(x, qw, qws, B);
  padw_kernel<<<(H + 255) / 256, 256, 0, stream>>>(fcw, fcw8, H);

  const int btiles = B / 16;                  // 2048
  const int wavesPerBlock = 8;                // 256 threads
  const int blocks = (btiles + wavesPerBlock - 1) / wavesPerBlock;
  resgemm_kernel<<<blocks, wavesPerBlock * 32, 0, stream>>>(qws, fcw8, hx, fcb, out, B, H);
}